// SINet_43997644980504
// MI455X (gfx1250) — compile-verified
//
#include <hip/hip_runtime.h>
#include <math.h>
#include <stdint.h>

typedef float v2f __attribute__((ext_vector_type(2)));
typedef float v8f __attribute__((ext_vector_type(8)));

#define NGROUP 32
#define NC     1024

// workspace layout (float offsets)
#define WS_POOLED 0
#define WS_CWO    (WS_POOLED + NGROUP*NC)
#define WS_XFM    (WS_CWO    + NGROUP*NC)
#define WS_CAL    (WS_XFM    + NGROUP*NC)
#define WS_KERN   (WS_CAL    + NGROUP*NC)          // [32][1024][16]
#define WS_W      (WS_KERN   + NGROUP*NC*16)       // [32][3][128]

__device__ __forceinline__ int iclamp(int v, int lo, int hi) {
  return v < lo ? lo : (v > hi ? hi : v);
}

// ---------------- K1: per-(g,c) reductions ----------------
__global__ void __launch_bounds__(256) k1_stats(const float* __restrict__ x,
                                                float* __restrict__ ws) {
  int task = blockIdx.x * 8 + (threadIdx.x >> 5);   // 0..32767 = g*1024+c
  int lane = threadIdx.x & 31;
  int g = task >> 10, c = task & 1023;
  int bi = g >> 1, gi = g & 1;
  const float* p = x + ((size_t)(bi * 1024 + c)) * 1024 + gi * 512;
  float s_all = 0.f, s_int = 0.f, s_xf = 0.f;
  for (int i = lane; i < 512; i += 32) {
    float v = p[i];
    s_all += v;
    int idx = i & 127, hh = idx >> 3, wx = idx & 7;
    if (hh >= 1 && hh <= 14 && wx >= 1 && wx <= 6) s_int += v;
    if (i < 128) s_xf += v;
  }
  for (int off = 16; off > 0; off >>= 1) {
    s_all += __shfl_down(s_all, off);
    s_int += __shfl_down(s_int, off);
    s_xf  += __shfl_down(s_xf,  off);
  }
  if (lane == 0) {
    ws[WS_POOLED + task] = s_all * (1.f / 512.f);
    ws[WS_CWO    + task] = s_int * (1.f / 336.f);
    ws[WS_XFM    + task] = s_xf  * (1.f / 128.f);
  }
}

// ---------------- K2: SE + center mask -> cal ----------------
__global__ void __launch_bounds__(128) k2_cal(const float* __restrict__ W1,
                                              const float* __restrict__ b1,
                                              const float* __restrict__ W2,
                                              const float* __restrict__ b2,
                                              float* __restrict__ ws) {
  __shared__ float red[3][128];
  int g = blockIdx.x, tid = threadIdx.x;
  float p0 = 0.f, p1 = 0.f, p2 = 0.f;
  for (int c = tid; c < 1024; c += 128) {
    float pv = ws[WS_POOLED + g * 1024 + c];
    p0 += W1[c] * pv;
    p1 += W1[1024 + c] * pv;
    p2 += W1[2048 + c] * pv;
  }
  red[0][tid] = p0; red[1][tid] = p1; red[2][tid] = p2;
  __syncthreads();
  for (int s = 64; s > 0; s >>= 1) {
    if (tid < s) {
      red[0][tid] += red[0][tid + s];
      red[1][tid] += red[1][tid + s];
      red[2][tid] += red[2][tid + s];
    }
    __syncthreads();
  }
  float s0 = fmaxf(red[0][0] + b1[0], 0.f);
  float s1 = fmaxf(red[1][0] + b1[1], 0.f);
  float s2 = fmaxf(red[2][0] + b1[2], 0.f);
  for (int c = tid; c < 1024; c += 128) {
    float z = W2[c * 3] * s0 + W2[c * 3 + 1] * s1 + W2[c * 3 + 2] * s2 + b2[c];
    float se = 1.f / (1.f + expf(-z));
    float center = ws[WS_CWO + g * 1024 + c] / (ws[WS_POOLED + g * 1024 + c] + 1e-8f);
    ws[WS_CAL + g * 1024 + c] = (center > 1.f) ? se : 0.f;
  }
}

// ---------------- K3: sa (WMMA) + L1 norms + kern (WMMA) ----------------
__global__ void __launch_bounds__(256) k3_kern(const float* __restrict__ x,
                                               const float* __restrict__ Wk,
                                               const float* __restrict__ bk,
                                               const float* __restrict__ bn_g,
                                               const float* __restrict__ bn_b,
                                               const float* __restrict__ bn_rm,
                                               const float* __restrict__ bn_rv,
                                               float* __restrict__ ws) {
  __shared__ float wk_s[9 * 1024];   // 36 KB
  __shared__ float cal_s[1024];      //  4 KB
  __shared__ float sa_s[9 * 128];    // 4.5 KB
  __shared__ float saB[128 * 16];    //  8 KB
  int g = blockIdx.x, tid = threadIdx.x;
  int bi = g >> 1, gi = g & 1;
  const float* xb = x + ((size_t)bi) * 1024 * 1024 + gi * 512;  // xf[c][n] = xb[c*1024+n]

  for (int i = tid; i < 9 * 1024; i += 256) wk_s[i] = Wk[i];
  for (int i = tid; i < 1024; i += 256)     cal_s[i] = ws[WS_CAL + g * 1024 + i];
  __syncthreads();

  int wave = tid >> 5, lane = tid & 31;
  int half = lane >> 4, lm = lane & 15;

  // --- step a: sa(16x128) = Wk_pad(16x1024) @ xf_se(1024x128), 8 waves = 8 N-tiles
  // Rows of D are independent, so A rows 9..15 may carry clamped garbage (row 8):
  // unconditional LDS loads, no EXEC divergence in the WMMA loop.
  {
    int rowi = (lm < 9) ? lm : 8;
    int n = wave * 16 + lm;
    v8f acc = {0.f, 0.f, 0.f, 0.f, 0.f, 0.f, 0.f, 0.f};
    for (int k = 0; k < 1024; k += 4) {
      int ka = k + half * 2;
      v2f a, b;
      a.x = wk_s[rowi * 1024 + ka];
      a.y = wk_s[rowi * 1024 + ka + 1];
      b.x = fmaxf(xb[(size_t)ka       * 1024 + n] * cal_s[ka],     0.f);
      b.y = fmaxf(xb[(size_t)(ka + 1) * 1024 + n] * cal_s[ka + 1], 0.f);
      acc = __builtin_amdgcn_wmma_f32_16x16x4_f32(false, a, false, b,
                                                  (short)0, acc, false, false);
    }
    for (int r = 0; r < 8; ++r) {
      int o = r + half * 8;                 // C/D layout: M = r + 8*(lane>=16)
      if (o < 9) {
        float v = acc[r] + bk[o];
        v = bn_g[o] * (v - bn_rm[o]) * rsqrtf(bn_rv[o] + 1e-5f) + bn_b[o];
        sa_s[o * 128 + n] = fmaxf(v, 0.f);
      }
    }
  }
  __syncthreads();

  // --- L1 norm over k (9) per position, then over n (128) per k
  if (tid < 128) {
    float s = 0.f;
    for (int o = 0; o < 9; ++o) s += fabsf(sa_s[o * 128 + tid]);
    float inv = 1.f / fmaxf(s, 1e-12f);
    for (int o = 0; o < 9; ++o) sa_s[o * 128 + tid] *= inv;
  }
  __syncthreads();
  if (tid < 9) {
    float s = 0.f;
    for (int n = 0; n < 128; ++n) s += fabsf(sa_s[tid * 128 + n]);
    float inv = 1.f / fmaxf(s, 1e-12f);
    for (int n = 0; n < 128; ++n) sa_s[tid * 128 + n] *= inv;
  }
  __syncthreads();
  for (int i = tid; i < 128 * 16; i += 256) {    // B-layout: saB[n][kcol], cols 9..15 = 0
    int n = i >> 4, o = i & 15;
    float vv = sa_s[((o < 9) ? o : 8) * 128 + n];  // unconditional load + select
    saB[i] = (o < 9) ? vv : 0.f;
  }
  __syncthreads();

  // --- step c: kern(1024x16) = xf_se(1024x128) @ saB(128x16); 64 M-tiles, 8 per wave
  for (int mt = wave * 8; mt < wave * 8 + 8; ++mt) {
    int crow = mt * 16 + lm;                // A layout: M = lane&15 in both halves
    float cl = cal_s[crow];
    const float* xr = xb + (size_t)crow * 1024;
    v8f acc = {0.f, 0.f, 0.f, 0.f, 0.f, 0.f, 0.f, 0.f};
    for (int k = 0; k < 128; k += 4) {
      int ka = k + half * 2;
      v2f a, b;
      a.x = fmaxf(xr[ka] * cl, 0.f);
      a.y = fmaxf(xr[ka + 1] * cl, 0.f);
      b.x = saB[ka * 16 + lm];
      b.y = saB[(ka + 1) * 16 + lm];
      acc = __builtin_amdgcn_wmma_f32_16x16x4_f32(false, a, false, b,
                                                  (short)0, acc, false, false);
    }
    // cols >= 9 multiplied against zero B columns -> exact 0; store unconditionally
    for (int r = 0; r < 8; ++r) {
      int cch = mt * 16 + r + half * 8;
      ws[WS_KERN + ((size_t)g * 1024 + cch) * 16 + lm] = acc[r] * (1.f / 9.f);
    }
  }
}

// ---------------- K4: 9-tap field + softmax + suppress -> w ----------------
// LDS staging of the raw xl tile via async global->LDS DMA (ASYNCcnt tracked);
// cal >= 0 so relu(v*cal) == cal*relu(v): fold cal per-channel at consume time.
__global__ void __launch_bounds__(256) k4_field(const float* __restrict__ x,
                                                float* __restrict__ ws) {
  __shared__ float xls[64 * 128];   // 32 KB chunk of raw xl
  __shared__ float cal_sh[64];
  __shared__ float rbuf[128];
  int g = blockIdx.x / 3, t = blockIdx.x % 3, tid = threadIdx.x;
  int bi = g >> 1, gi = g & 1;
  const float* xb = x + ((size_t)bi) * 1024 * 1024 + gi * 512;
  int pos = tid & 127;
  int y = pos >> 3, xw = pos & 7;
  float facc = 0.f;
  for (int c0 = 0; c0 < 1024; c0 += 64) {
    // stage: 64 channels x 128 floats, 16B per async op per lane
    for (int i = tid; i < 64 * 32; i += 256) {
      int cc = i >> 5, q = i & 31;                       // q: float4 index in row
      uint32_t ldsoff = (uint32_t)(uintptr_t)(xls + cc * 128 + q * 4);
      uint64_t gsrc = (uint64_t)(uintptr_t)(xb + (size_t)(c0 + cc) * 1024 +
                                            (t + 1) * 128 + q * 4);
      asm volatile("global_load_async_to_lds_b128 %0, %1, off"
                   :: "v"(ldsoff), "v"(gsrc) : "memory");
    }
    if (tid < 64) cal_sh[tid] = ws[WS_CAL + g * 1024 + c0 + tid];
    if (c0 + 64 < 1024)
      __builtin_prefetch(xb + (size_t)(c0 + 64) * 1024 + (t + 1) * 128, 0, 1);
    asm volatile("s_wait_asynccnt 0x0" ::: "memory");
    __syncthreads();
    if (tid < 128) {
      for (int cc = 0; cc < 64; ++cc) {
        const float* kr = ws + WS_KERN + ((size_t)g * 1024 + c0 + cc) * 16;
        const float* xp = xls + cc * 128;
        float a2 = 0.f;
#pragma unroll
        for (int i = 0; i < 3; ++i) {
          int yy = iclamp(y + i - 1, 0, 15);
#pragma unroll
          for (int j = 0; j < 3; ++j) {
            int xx = iclamp(xw + j - 1, 0, 7);
            a2 += fmaxf(xp[yy * 8 + xx], 0.f) * kr[i * 3 + j];
          }
        }
        facc += a2 * cal_sh[cc];
      }
    }
    __syncthreads();   // before next chunk overwrites xls
  }
  if (tid < 128) rbuf[tid] = facc;
  __syncthreads();
  for (int s = 64; s > 0; s >>= 1) {
    if (tid < s) rbuf[tid] = fmaxf(rbuf[tid], rbuf[tid + s]);
    __syncthreads();
  }
  float m = rbuf[0];
  __syncthreads();
  float e = expf(facc - m);
  if (tid < 128) rbuf[tid] = e;
  __syncthreads();
  for (int s = 64; s > 0; s >>= 1) {
    if (tid < s) rbuf[tid] += rbuf[tid + s];
    __syncthreads();
  }
  float tot = rbuf[0];
  if (tid < 128) {
    float pr = fmaxf(e / tot, 1e-4f);
    float a = 1.f / (pr * 128.f);
    float wv = (a < 1.f) ? expf((a - 1.f) * 5.f) : 1.f;
    ws[WS_W + (g * 3 + t) * 128 + pos] = wv;
  }
}

// ---------------- K5: final streaming pass ----------------
__global__ void __launch_bounds__(256) k5_out(const float* __restrict__ x,
                                              const float* __restrict__ ws,
                                              float* __restrict__ out) {
  int task = blockIdx.x * 8 + (threadIdx.x >> 5);
  int lane = threadIdx.x & 31;
  int g = task >> 10, c = task & 1023;
  int bi = g >> 1, gi = g & 1;
  size_t base = ((size_t)(bi * 1024 + c)) * 1024 + gi * 512;
  const float* p = x + base;
  float* o = out + base;
  const float* wg = ws + WS_W + g * 384;
  float v[16];
  float num = 0.f, den = 0.f;
#pragma unroll
  for (int it = 0; it < 16; ++it) {
    int idx = lane + it * 32;
    float val = p[idx];
    v[it] = val;
    if (idx >= 128) {
      float om = 1.f - wg[idx - 128];
      num += om * val;
      den += om;
    }
  }
#pragma unroll
  for (int off = 1; off < 32; off <<= 1) {
    num += __shfl_xor(num, off);
    den += __shfl_xor(den, off);
  }
  float addf = 0.1f * num / den;              // 0.1*mean((1-f)*xl)/mean(1-f)
  float xfm01 = 0.1f * ws[WS_XFM + task];
#pragma unroll
  for (int it = 0; it < 16; ++it) {
    int idx = lane + it * 32;
    float r = (idx < 128) ? (v[it] + addf)
                          : (v[it] * wg[idx - 128] + xfm01);
    o[idx] = r;
  }
}

extern "C" void kernel_launch(void* const* d_in, const int* in_sizes, int n_in,
                              void* d_out, int out_size, void* d_ws, size_t ws_size,
                              hipStream_t stream) {
  (void)in_sizes; (void)n_in; (void)out_size; (void)ws_size;
  const float* x    = (const float*)d_in[0];
  const float* Wk   = (const float*)d_in[1];
  const float* bk   = (const float*)d_in[2];
  const float* bng  = (const float*)d_in[3];
  const float* bnb  = (const float*)d_in[4];
  const float* bnrm = (const float*)d_in[5];
  const float* bnrv = (const float*)d_in[6];
  const float* W1   = (const float*)d_in[7];
  const float* b1   = (const float*)d_in[8];
  const float* W2   = (const float*)d_in[9];
  const float* b2   = (const float*)d_in[10];
  float* out = (float*)d_out;
  float* ws  = (float*)d_ws;

  k1_stats<<<4096, 256, 0, stream>>>(x, ws);
  k2_cal  <<<32,   128, 0, stream>>>(W1, b1, W2, b2, ws);
  k3_kern <<<32,   256, 0, stream>>>(x, Wk, bk, bng, bnb, bnrm, bnrv, ws);
  k4_field<<<96,   256, 0, stream>>>(x, ws);
  k5_out  <<<4096, 256, 0, stream>>>(x, ws, out);
}